// GGNNModel_59425167507917
// MI455X (gfx1250) — compile-verified
//
#include <hip/hip_runtime.h>
#include <hip/hip_bf16.h>

// ---------------------------------------------------------------------------
// GGNN on MI455X (gfx1250).  fp32 WMMA (v_wmma_f32_16x16x4_f32) for all GEMMs.
// v3: Ncols is a template parameter -> B-fragment addresses become immediate
// offsets on global_load_b32 (kills the v_mul_u64/v_add_u64 chains seen in
// v2's inner loop); 32-row block tile so each B fragment feeds two WMMAs.
// Scratch usage ~193M floats (~0.77 GB) in d_ws.
// ---------------------------------------------------------------------------

typedef __attribute__((ext_vector_type(2))) float v2f;
typedef __attribute__((ext_vector_type(8))) float v8f;

#define Nn   100000
#define Ee   1600000
#define Hh   128
#define Ll   4
#define Gg   512
#define INn  128
#define OUTn 16
#define HSn  64

// ---------------------------------------------------------------------------
// Generic WMMA GEMM: C[M,NC] = A[M,K] @ B[K,NC] (+bias) (+relu)
// Requirements: M % 32 == 0, K % 64 == 0, NC % 16 == 0 (compile-time).
// B is row-major [K, NC] (weights pre-transposed -> coalesced fragment loads).
// Block = 256 threads = 8 waves; block tile = 32 rows x min(NC,128) cols; wave
// j owns two 16x16 tiles (rows r..r+15, r+16..r+31) sharing each B fragment.
// A chunk (32 x 64) staged in LDS with pitch 68 so the stride-68 fragment
// reads hit distinct banks.
// ---------------------------------------------------------------------------
template <int NC>
__global__ __launch_bounds__(256)
void wmma_gemm_f32(const float* __restrict__ A, const float* __restrict__ B,
                   const float* __restrict__ bias, float* __restrict__ C,
                   int M, int K, int relu)
{
    __shared__ float As[32][68];
    const int row0 = blockIdx.x * 32;
    const int wave = threadIdx.x >> 5;
    const int lane = threadIdx.x & 31;
    const int half = lane >> 4;     // 0: lanes 0-15, 1: lanes 16-31
    const int lm   = lane & 15;
    const int col0 = blockIdx.y * 128 + wave * 16;
    const bool active = (col0 < NC);   // wave-uniform -> EXEC all-ones in WMMA
                                       // (always true when NC >= 128)
    v8f acc0 = {};
    v8f acc1 = {};

    for (int kc = 0; kc < K; kc += 64) {
        // stage A[row0:row0+32, kc:kc+64] into LDS (two float4 per thread)
#pragma unroll
        for (int s = 0; s < 2; ++s) {
            const int slot = threadIdx.x + s * 256;  // 0..511
            const int r  = slot >> 4;                // 0..31
            const int f4 = slot & 15;                // 0..15
            const float4 v = *(const float4*)(A + (size_t)(row0 + r) * K + kc + f4 * 4);
            As[r][f4 * 4 + 0] = v.x;
            As[r][f4 * 4 + 1] = v.y;
            As[r][f4 * 4 + 2] = v.z;
            As[r][f4 * 4 + 3] = v.w;
        }
        __syncthreads();

        if (active) {
            // per-chunk base; kk*NC folds into the load immediate offsets
            const float* bbase = B + (size_t)(kc + 2 * half) * NC + col0 + lm;
#pragma unroll
            for (int kk = 0; kk < 64; kk += 4) {
                // A fragment (ISA 7.12.2): VGPR0 = K {0 | 2}, VGPR1 = K {1 | 3}
                v2f a0, a1, b;
                a0.x = As[lm][kk + 2 * half + 0];
                a0.y = As[lm][kk + 2 * half + 1];
                a1.x = As[16 + lm][kk + 2 * half + 0];
                a1.y = As[16 + lm][kk + 2 * half + 1];
                // B fragment: VGPR0 = rows {k | k+2}, VGPR1 = rows {k+1 | k+3}
                b.x = bbase[kk * NC];
                b.y = bbase[(kk + 1) * NC];
                acc0 = __builtin_amdgcn_wmma_f32_16x16x4_f32(
                        false, a0, false, b, (short)0, acc0, false, false);
                acc1 = __builtin_amdgcn_wmma_f32_16x16x4_f32(
                        false, a1, false, b, (short)0, acc1, false, false);
            }
        }
        __syncthreads();
    }

    if (active) {
        const float bv = bias ? bias[col0 + lm] : 0.0f;
#pragma unroll
        for (int v = 0; v < 8; ++v) {
            const int rm = v + 8 * half;          // C layout: VGPR v -> rows v / v+8
            float v0 = acc0[v] + bv;
            float v1 = acc1[v] + bv;
            if (relu) { v0 = fmaxf(v0, 0.0f); v1 = fmaxf(v1, 0.0f); }
            C[(size_t)(row0 + rm)      * NC + col0 + lm] = v0;
            C[(size_t)(row0 + 16 + rm) * NC + col0 + lm] = v1;
        }
    }
}

// ---------------------------------------------------------------------------
// Simple transpose: WT[C,R] = W[R,C]^T
// ---------------------------------------------------------------------------
__global__ void transpose_k(const float* __restrict__ W, float* __restrict__ WT,
                            int R, int Ccols)
{
    int i = blockIdx.x * blockDim.x + threadIdx.x;
    if (i >= R * Ccols) return;
    int r = i / Ccols, c = i % Ccols;
    WT[(size_t)c * R + r] = W[i];
}

__global__ void fill_f(float* __restrict__ p, float v, size_t n)
{
    size_t i = (size_t)blockIdx.x * blockDim.x + threadIdx.x;
    if (i < n) p[i] = v;
}

__global__ void fill_u(unsigned* __restrict__ p, unsigned v, size_t n)
{
    size_t i = (size_t)blockIdx.x * blockDim.x + threadIdx.x;
    if (i < n) p[i] = v;
}

// ---------------------------------------------------------------------------
// Edge scatter-add: agg[dst] += m[src]   (one thread per (edge, float4))
// Source/target arrays (51 MB each) are L2-resident -> atomics stay on-chip.
// ---------------------------------------------------------------------------
__global__ void edge_scatter(const float* __restrict__ m, const int* __restrict__ src,
                             const int* __restrict__ dst, float* __restrict__ agg)
{
    long long i = (long long)blockIdx.x * blockDim.x + threadIdx.x;
    const long long total = (long long)Ee * (Hh / 4);
    if (i >= total) return;
    int e  = (int)(i >> 5);         // Hh/4 == 32
    int c4 = (int)(i & 31);
    int s = src[e], d = dst[e];
    const float4 v = *(const float4*)(m + (size_t)s * Hh + c4 * 4);
    float* p = agg + (size_t)d * Hh + c4 * 4;
    atomicAdd(p + 0, v.x);
    atomicAdd(p + 1, v.y);
    atomicAdd(p + 2, v.z);
    atomicAdd(p + 3, v.w);
}

// ---------------------------------------------------------------------------
// GRU elementwise: x = (1-z)*n + z*x  given gi = agg@Wih^T+bih, gh = x@Whh^T+bhh
// ---------------------------------------------------------------------------
__global__ void gru_elem(const float* __restrict__ gi, const float* __restrict__ gh,
                         float* __restrict__ x)
{
    size_t i = (size_t)blockIdx.x * blockDim.x + threadIdx.x;
    if (i >= (size_t)Nn * Hh) return;
    size_t n = i / Hh;
    int    c = (int)(i % Hh);
    const float* gin = gi + n * (3 * Hh);
    const float* ghn = gh + n * (3 * Hh);
    float ir = gin[c], iz = gin[Hh + c], in_ = gin[2 * Hh + c];
    float hr = ghn[c], hz = ghn[Hh + c], hn  = ghn[2 * Hh + c];
    float r  = 1.0f / (1.0f + expf(-(ir + hr)));
    float z  = 1.0f / (1.0f + expf(-(iz + hz)));
    float nv = tanhf(in_ + r * hn);
    x[i] = (1.0f - z) * nv + z * x[i];
}

// cat[:, blk*H:(blk+1)*H] = x + h0
__global__ void residual_cat(const float* __restrict__ x, const float* __restrict__ h0,
                             float* __restrict__ catb, int blk)
{
    size_t i = (size_t)blockIdx.x * blockDim.x + threadIdx.x;
    if (i >= (size_t)Nn * Hh) return;
    size_t n = i / Hh;
    int    c = (int)(i % Hh);
    catb[n * (size_t)(Ll * Hh) + (size_t)blk * Hh + c] = x[i] + h0[i];
}

// gate[n] = hidden[n,:] . gate_w + gate_b
__global__ void gate_dot(const float* __restrict__ hidden, const float* __restrict__ gw,
                         const float* __restrict__ gb, float* __restrict__ gate)
{
    int n = blockIdx.x * blockDim.x + threadIdx.x;
    if (n >= Nn) return;
    float s = gb[0];
    const float* h = hidden + (size_t)n * Hh;
#pragma unroll 4
    for (int c = 0; c < Hh; ++c) s += h[c] * gw[c];
    gate[n] = s;
}

// float ordering trick for atomicMax on unsigned
__device__ __forceinline__ unsigned f2ord(float f)
{
    unsigned u = __float_as_uint(f);
    return (u & 0x80000000u) ? ~u : (u | 0x80000000u);
}
__device__ __forceinline__ float ord2f(unsigned u)
{
    return (u & 0x80000000u) ? __uint_as_float(u & 0x7fffffffu) : __uint_as_float(~u);
}

__global__ void seg_max(const float* __restrict__ gate, const int* __restrict__ batch,
                        unsigned* __restrict__ gmax_ord)
{
    int n = blockIdx.x * blockDim.x + threadIdx.x;
    if (n >= Nn) return;
    atomicMax(&gmax_ord[batch[n]], f2ord(gate[n]));
}

// ex[n] = exp(gate[n] - gmax[batch[n]]);  denom[b] += ex  (ex overwrites gate)
__global__ void softmax_ex(float* __restrict__ gate, const int* __restrict__ batch,
                           const unsigned* __restrict__ gmax_ord, float* __restrict__ denom)
{
    int n = blockIdx.x * blockDim.x + threadIdx.x;
    if (n >= Nn) return;
    int b = batch[n];
    float e = expf(gate[n] - ord2f(gmax_ord[b]));
    gate[n] = e;
    atomicAdd(&denom[b], e);
}

// pooled[b] += (ex[n]/denom[b]) * hidden[n]
__global__ void pool_k(const float* __restrict__ hidden, const float* __restrict__ ex,
                       const float* __restrict__ denom, const int* __restrict__ batch,
                       float* __restrict__ pooled)
{
    long long i = (long long)blockIdx.x * blockDim.x + threadIdx.x;
    const long long total = (long long)Nn * (Hh / 4);
    if (i >= total) return;
    size_t n  = (size_t)(i >> 5);
    int    c4 = (int)(i & 31);
    int b = batch[n];
    float alpha = ex[n] / denom[b];
    const float4 hv = *(const float4*)(hidden + n * Hh + c4 * 4);
    float* p = pooled + (size_t)b * Hh + c4 * 4;
    atomicAdd(p + 0, alpha * hv.x);
    atomicAdd(p + 1, alpha * hv.y);
    atomicAdd(p + 2, alpha * hv.z);
    atomicAdd(p + 3, alpha * hv.w);
}

// ---------------------------------------------------------------------------
extern "C" void kernel_launch(void* const* d_in, const int* in_sizes, int n_in,
                              void* d_out, int out_size, void* d_ws, size_t ws_size,
                              hipStream_t stream)
{
    (void)in_sizes; (void)n_in; (void)out_size; (void)ws_size;

    const float* node_embed = (const float*)d_in[0];
    const int*   edge_index = (const int*)  d_in[1];
    const int*   batch      = (const int*)  d_in[2];
    const float* node_w     = (const float*)d_in[3];
    const float* node_b     = (const float*)d_in[4];
    const float* ggc_w      = (const float*)d_in[5];   // [L,L,H,H]
    const float* gru_w_ih   = (const float*)d_in[6];   // [L,3H,H]
    const float* gru_w_hh   = (const float*)d_in[7];   // [L,3H,H]
    const float* gru_b_ih   = (const float*)d_in[8];   // [L,3H]
    const float* gru_b_hh   = (const float*)d_in[9];   // [L,3H]
    const float* cat_w      = (const float*)d_in[10];  // [H,4H]
    const float* cat_b      = (const float*)d_in[11];
    const float* gate_w     = (const float*)d_in[12];  // [1,H]
    const float* gate_b     = (const float*)d_in[13];
    const float* mlp_w1     = (const float*)d_in[14];  // [HS,H]
    const float* mlp_b1     = (const float*)d_in[15];
    const float* mlp_w2     = (const float*)d_in[16];  // [HS,HS]
    const float* mlp_b2     = (const float*)d_in[17];
    const float* out_w      = (const float*)d_in[18];  // [OUT,HS]
    const float* out_b      = (const float*)d_in[19];
    const float* line_w     = (const float*)d_in[20];  // [OUT,H]
    const float* line_b     = (const float*)d_in[21];

    float* outp    = (float*)d_out;          // [G,OUT] first
    float* lineout = outp + Gg * OUTn;       // then [N,OUT]

    const int* srcIdx = edge_index;          // edge_index[0,:]
    const int* dstIdx = edge_index + Ee;     // edge_index[1,:]

    // ---- scratch layout ----------------------------------------------------
    float* ws = (float*)d_ws;
    size_t off = 0;
    auto alloc = [&](size_t nfloats) { float* p = ws + off; off += nfloats; return p; };
    float* h0     = alloc((size_t)Nn * Hh);
    float* x      = alloc((size_t)Nn * Hh);
    float* m      = alloc((size_t)Nn * Hh);
    float* agg    = alloc((size_t)Nn * Hh);
    float* gi     = alloc((size_t)Nn * 3 * Hh);
    float* gh     = alloc((size_t)Nn * 3 * Hh);
    float* catb   = alloc((size_t)Nn * Ll * Hh);
    float* hidden = alloc((size_t)Nn * Hh);
    float* gate   = alloc((size_t)Nn);
    unsigned* gmax_ord = (unsigned*)alloc(Gg);
    float* denom  = alloc(Gg);
    float* pooled = alloc((size_t)Gg * Hh);
    float* mb1    = alloc((size_t)Gg * HSn);
    float* mb2    = alloc((size_t)Gg * HSn);
    float* node_wT = alloc((size_t)INn * Hh);
    float* w_ihT   = alloc((size_t)Ll * Hh * 3 * Hh);
    float* w_hhT   = alloc((size_t)Ll * Hh * 3 * Hh);
    float* cat_wT  = alloc((size_t)Ll * Hh * Hh);
    float* mlp1T   = alloc((size_t)Hh * HSn);
    float* mlp2T   = alloc((size_t)HSn * HSn);
    float* outT    = alloc((size_t)HSn * OUTn);
    float* lineT   = alloc((size_t)Hh * OUTn);

    const int T = 256;
    auto blks = [&](size_t n) { return (unsigned)((n + T - 1) / T); };

    auto gemm = [&](const float* A, const float* Bm, const float* bias, float* C,
                    int M, int K, int Ncols, int relu) {
        dim3 g(M / 32, (Ncols + 127) / 128);
        switch (Ncols) {
        case 128: wmma_gemm_f32<128><<<g, 256, 0, stream>>>(A, Bm, bias, C, M, K, relu); break;
        case 384: wmma_gemm_f32<384><<<g, 256, 0, stream>>>(A, Bm, bias, C, M, K, relu); break;
        case 64:  wmma_gemm_f32<64> <<<g, 256, 0, stream>>>(A, Bm, bias, C, M, K, relu); break;
        case 16:  wmma_gemm_f32<16> <<<g, 256, 0, stream>>>(A, Bm, bias, C, M, K, relu); break;
        }
    };

    // ---- weight transposes (W^T so GEMM B is [K,N] row-major) --------------
    transpose_k<<<blks(Hh * INn), T, 0, stream>>>(node_w, node_wT, Hh, INn);
    for (int b = 0; b < Ll; ++b) {
        transpose_k<<<blks(3 * Hh * Hh), T, 0, stream>>>(
            gru_w_ih + (size_t)b * 3 * Hh * Hh, w_ihT + (size_t)b * Hh * 3 * Hh, 3 * Hh, Hh);
        transpose_k<<<blks(3 * Hh * Hh), T, 0, stream>>>(
            gru_w_hh + (size_t)b * 3 * Hh * Hh, w_hhT + (size_t)b * Hh * 3 * Hh, 3 * Hh, Hh);
    }
    transpose_k<<<blks(Hh * Ll * Hh), T, 0, stream>>>(cat_w, cat_wT, Hh, Ll * Hh);
    transpose_k<<<blks(HSn * Hh), T, 0, stream>>>(mlp_w1, mlp1T, HSn, Hh);
    transpose_k<<<blks(HSn * HSn), T, 0, stream>>>(mlp_w2, mlp2T, HSn, HSn);
    transpose_k<<<blks(OUTn * HSn), T, 0, stream>>>(out_w, outT, OUTn, HSn);
    transpose_k<<<blks(OUTn * Hh), T, 0, stream>>>(line_w, lineT, OUTn, Hh);

    // ---- h0 = node_embed @ node_w.T + node_b -------------------------------
    gemm(node_embed, node_wT, node_b, h0, Nn, INn, Hh, 0);

    // ---- residual GGC blocks ----------------------------------------------
    const size_t NH = (size_t)Nn * Hh;
    for (int blk = 0; blk < Ll; ++blk) {
        hipMemcpyAsync(x, h0, NH * sizeof(float), hipMemcpyDeviceToDevice, stream);
        for (int step = 0; step < Ll; ++step) {
            // m = x @ W[blk][step]   (W already [K=H, N=H] row-major)
            gemm(x, ggc_w + (size_t)(blk * Ll + step) * Hh * Hh, nullptr, m, Nn, Hh, Hh, 0);
            // agg = segment_sum(m[src] -> dst)
            fill_f<<<blks(NH), T, 0, stream>>>(agg, 0.0f, NH);
            edge_scatter<<<blks((size_t)Ee * (Hh / 4)), T, 0, stream>>>(m, srcIdx, dstIdx, agg);
            // GRU: gi = agg @ w_ih^T + b_ih ; gh = x @ w_hh^T + b_hh
            gemm(agg, w_ihT + (size_t)blk * Hh * 3 * Hh, gru_b_ih + (size_t)blk * 3 * Hh,
                 gi, Nn, Hh, 3 * Hh, 0);
            gemm(x,   w_hhT + (size_t)blk * Hh * 3 * Hh, gru_b_hh + (size_t)blk * 3 * Hh,
                 gh, Nn, Hh, 3 * Hh, 0);
            gru_elem<<<blks(NH), T, 0, stream>>>(gi, gh, x);
        }
        residual_cat<<<blks(NH), T, 0, stream>>>(x, h0, catb, blk);
    }

    // ---- hidden = cat(outs) @ cat_w.T + cat_b ------------------------------
    gemm(catb, cat_wT, cat_b, hidden, Nn, Ll * Hh, Hh, 0);

    // ---- lineout head ------------------------------------------------------
    gemm(hidden, lineT, line_b, lineout, Nn, Hh, OUTn, 0);

    // ---- attentional aggregation ------------------------------------------
    gate_dot<<<blks(Nn), T, 0, stream>>>(hidden, gate_w, gate_b, gate);
    fill_u<<<blks(Gg), T, 0, stream>>>(gmax_ord, 0u, Gg);
    seg_max<<<blks(Nn), T, 0, stream>>>(gate, batch, gmax_ord);
    fill_f<<<blks(Gg), T, 0, stream>>>(denom, 0.0f, Gg);
    softmax_ex<<<blks(Nn), T, 0, stream>>>(gate, batch, gmax_ord, denom);
    fill_f<<<blks((size_t)Gg * Hh), T, 0, stream>>>(pooled, 0.0f, (size_t)Gg * Hh);
    pool_k<<<blks((size_t)Nn * (Hh / 4)), T, 0, stream>>>(hidden, gate, denom, batch, pooled);

    // ---- MLP + out head ----------------------------------------------------
    gemm(pooled, mlp1T, mlp_b1, mb1, Gg, Hh, HSn, 1);
    gemm(mb1,    mlp2T, mlp_b2, mb2, Gg, HSn, HSn, 1);
    gemm(mb2,    outT,  out_b,  outp, Gg, HSn, OUTn, 0);
}